// TokenCoder_9345848836381
// MI455X (gfx1250) — compile-verified
//
#include <hip/hip_runtime.h>

// Native 4-wide float vector so __builtin_nontemporal_load/store lower to
// single global_load_b128 / global_store_b128 with TH=NT cache hints.
typedef float v4f __attribute__((ext_vector_type(4)));

#define N_TYPES 8
#define DIM 16
#define TABLE (N_TYPES * DIM)   // 128 entries per table
#define THREADS 256             // 8 wave32 waves per block
#define VEC_PER_THREAD 4        // grid-stride iterations per thread (amortize table setup)

// decoded = round((clamp(x, s, e) - s) / res) * res + s,  res = (e - s) / (size - 1)
__global__ __launch_bounds__(THREADS)
void tokencoder_roundtrip_kernel(const float* __restrict__ tks,
                                 const int*   __restrict__ tk_ids,
                                 const float* __restrict__ starts,
                                 const float* __restrict__ ends,
                                 const float* __restrict__ sizes,
                                 float*       __restrict__ out,
                                 long long nvec /* # of float4 chunks */) {
    // Per-(type,dim) parameter tables, laid out as 32 x float4 so the per-element
    // fetch is a single ds_load_b128 per table (4 lanes of one token broadcast).
    __shared__ v4f s_tab  [TABLE / 4];
    __shared__ v4f e_tab  [TABLE / 4];
    __shared__ v4f res_tab[TABLE / 4];
    __shared__ v4f inv_tab[TABLE / 4];

    const int t = threadIdx.x;
    if (t < TABLE) {
        // One IEEE divide per table entry here instead of 33.5M per-element divides.
        const float s   = starts[t];
        const float e   = ends[t];
        const float sz  = sizes[t];
        const float res = (e - s) / (sz - 1.0f);
        ((float*)s_tab)[t]   = s;
        ((float*)e_tab)[t]   = e;
        ((float*)res_tab)[t] = res;
        ((float*)inv_tab)[t] = 1.0f / res;
    }
    __syncthreads();

    const long long stride = (long long)blockDim.x * (long long)gridDim.x;
    const v4f* __restrict__ tks4 = (const v4f*)tks;
    v4f* __restrict__ out4       = (v4f*)out;

    for (long long i = (long long)blockIdx.x * blockDim.x + t; i < nvec; i += stride) {
        // Speculative prefetch of next grid-stride chunk (global_prefetch_b8);
        // translation failures are silently dropped, so no bounds check needed.
        __builtin_prefetch((const void*)(tks4 + i + stride), 0, 0);

        const long long tok  = i >> 2;          // token index (DIM=16 -> 4 vec4/token)
        const int       part = (int)(i & 3);    // which 4-dim slice of the token
        const int       id   = tk_ids[tok];     // 4 lanes share one word -> broadcast
        const int       tidx = (id << 2) + part;

        // Streaming input: read-once, 128 MiB -> non-temporal (don't thrash 192MB L2).
        const v4f x  = __builtin_nontemporal_load(tks4 + i);
        const v4f s  = s_tab[tidx];
        const v4f e  = e_tab[tidx];
        const v4f r  = res_tab[tidx];
        const v4f ir = inv_tab[tidx];

        v4f d;
#pragma unroll
        for (int k = 0; k < 4; ++k) {
            const float xc = fminf(fmaxf(x[k], s[k]), e[k]);          // clamp to [s, e]
            const float q  = __builtin_rintf((xc - s[k]) * ir[k]);    // v_rndne (RNE)
            d[k]           = __builtin_fmaf(q, r[k], s[k]);           // dequantize
        }

        // Streaming output: write-once, 128 MiB -> non-temporal store.
        __builtin_nontemporal_store(d, out4 + i);
    }
}

extern "C" void kernel_launch(void* const* d_in, const int* in_sizes, int n_in,
                              void* d_out, int out_size, void* d_ws, size_t ws_size,
                              hipStream_t stream) {
    (void)n_in; (void)out_size; (void)d_ws; (void)ws_size;

    const float* tks    = (const float*)d_in[0];  // [B, T, DIM] f32
    const int*   tk_ids = (const int*)  d_in[1];  // [B, T] i32
    const float* starts = (const float*)d_in[2];  // [N_TYPES, DIM] f32
    const float* ends   = (const float*)d_in[3];  // [N_TYPES, DIM] f32
    const float* sizes  = (const float*)d_in[4];  // [N_TYPES, DIM] f32
    float*       out    = (float*)d_out;          // [B, T, DIM] f32

    const long long n    = (long long)in_sizes[0];   // 33,554,432 floats
    const long long nvec = n >> 2;                   // float4 chunks (DIM=16 -> exact)

    const long long per_block = (long long)THREADS * VEC_PER_THREAD;
    long long blocks = (nvec + per_block - 1) / per_block;   // 8192 for the ref shape
    if (blocks < 1) blocks = 1;

    tokencoder_roundtrip_kernel<<<(dim3)(unsigned)blocks, THREADS, 0, stream>>>(
        tks, tk_ids, starts, ends, sizes, out, nvec);
}